// SelectionLayer_10505490006258
// MI455X (gfx1250) — compile-verified
//
#include <hip/hip_runtime.h>
#include <stdint.h>

typedef float v2f __attribute__((ext_vector_type(2)));
typedef float v8f __attribute__((ext_vector_type(8)));

#define BATCH 8
#define NROW 16384
#define CDIM 256
#define TOPN 2048
#define TOTROWS (BATCH * NROW)        // 131072
#define ROWS_PER_BLOCK 128            // 8 waves x 16 rows
#define K1_BLOCKS (TOTROWS / ROWS_PER_BLOCK)  // 1024

// ---------------- K1: scores via V_WMMA_F32_16X16X4_F32 + block partial sums ----
__global__ __launch_bounds__(256)
void score_wmma_kernel(const float* __restrict__ x_in,
                       const float* __restrict__ w,
                       const float* __restrict__ bias,
                       float* __restrict__ scores,
                       float* __restrict__ psum,
                       float* __restrict__ psq) {
    __shared__ float ssum[16];
    __shared__ float ssq[16];
    const int tid  = threadIdx.x;
    const int wave = tid >> 5;
    const int lane = tid & 31;
    const int m    = lane & 15;     // row within 16-row tile
    const int half = lane >> 4;     // 0 -> K pair (0,1), 1 -> K pair (2,3)
    const int rowbase = blockIdx.x * ROWS_PER_BLOCK + wave * 16;
    const float* rowp = x_in + (size_t)(rowbase + m) * CDIM;

    v8f c = {};
    // A: 16x4 f32 tile (lane m holds {k+2*half, k+2*half+1}); B: w chunk
    // broadcast across all 16 columns -> every D column equals the row score.
    for (int k0 = 0; k0 < CDIM; k0 += 4) {
        const int ko = k0 + 2 * half;
        v2f a  = *(const v2f*)(rowp + ko);
        v2f bb = *(const v2f*)(w + ko);
        c = __builtin_amdgcn_wmma_f32_16x16x4_f32(false, a, false, bb,
                                                  (short)0, c, false, false);
    }
    const float bv = bias[0];
    // Lane 0 holds rows 0..7 (N=0 column), lane 16 holds rows 8..15.
    if (m == 0) {
        float s = 0.f, q = 0.f;
        const int base = rowbase + half * 8;
        #pragma unroll
        for (int i = 0; i < 8; ++i) {
            float sc = c[i] + bv;
            scores[base + i] = sc;
            s += sc;
            q += sc * sc;
        }
        ssum[wave * 2 + half] = s;
        ssq [wave * 2 + half] = q;
    }
    __syncthreads();
    if (tid == 0) {   // fixed-order -> deterministic
        float s = 0.f, q = 0.f;
        for (int i = 0; i < 16; ++i) { s += ssum[i]; q += ssq[i]; }
        psum[blockIdx.x] = s;
        psq [blockIdx.x] = q;
    }
}

// ---------------- K2: final deterministic tree reduction -> mean, 1/sqrt(var+eps)
__global__ __launch_bounds__(512)
void reduce_kernel(const float* __restrict__ psum,
                   const float* __restrict__ psq,
                   float* __restrict__ stats) {
    __shared__ float s1[512];
    __shared__ float s2[512];
    const int t = threadIdx.x;
    s1[t] = psum[t] + psum[t + 512];
    s2[t] = psq [t] + psq [t + 512];
    __syncthreads();
    for (int off = 256; off > 0; off >>= 1) {
        if (t < off) { s1[t] += s1[t + off]; s2[t] += s2[t + off]; }
        __syncthreads();
    }
    if (t == 0) {
        const float inv = 1.0f / (float)TOTROWS;
        float mean = s1[0] * inv;
        float var  = fmaxf(s2[0] * inv - mean * mean, 0.0f);
        stats[0] = mean;
        stats[1] = rsqrtf(var + 1e-5f);
    }
}

// ---------------- K3: per-batch 64-bit-key bitonic sort in LDS (128 KB) --------
__global__ __launch_bounds__(1024)
void sort_kernel(const float* __restrict__ scores,
                 const float* __restrict__ stats,
                 const float* __restrict__ gamma,
                 const float* __restrict__ beta,
                 unsigned* __restrict__ idxout) {
    extern __shared__ unsigned long long keys[];   // NROW * 8 = 131072 bytes
    const int b   = blockIdx.x;
    const int tid = threadIdx.x;
    const float mean = stats[0], istd = stats[1];
    const float g = gamma[0], be = beta[0];

    for (int i = tid; i < NROW; i += 1024) {
        float s = scores[b * NROW + i];
        float v = (s - mean) * istd * g + be;        // BN (batch stats)
        float r = fmaxf(v, 0.0f);                    // ReLU: r >= 0
        // non-negative float bits are monotone as uint; low bits = stable index
        keys[i] = ((unsigned long long)__float_as_uint(r) << 32) | (unsigned)i;
    }
    __syncthreads();

    for (unsigned k = 2; k <= (unsigned)NROW; k <<= 1) {
        for (unsigned j = k >> 1; j > 0; j >>= 1) {
            for (int t = tid; t < NROW / 2; t += 1024) {
                unsigned tt = (unsigned)t;
                unsigned i  = 2u * tt - (tt & (j - 1u));  // bit j of i clear
                unsigned p  = i + j;
                unsigned long long a = keys[i];
                unsigned long long c = keys[p];
                bool asc = ((i & k) == 0u);
                if ((a > c) == asc) { keys[i] = c; keys[p] = a; }
            }
            __syncthreads();
        }
    }

    for (int t = tid; t < TOPN; t += 1024)
        idxout[b * TOPN + t] = (unsigned)(keys[t] & 0xffffffffull);
}

// ---------------- K4: gather selected rows of x_select (float4, 1 block/row) ---
__global__ __launch_bounds__(64)
void gather_kernel(const float* __restrict__ x_select,
                   const unsigned* __restrict__ idx,
                   float* __restrict__ out) {
    const int row = blockIdx.x;          // 0 .. BATCH*TOPN-1
    const int b   = row >> 11;           // / TOPN
    const unsigned gidx = idx[row];
    const float4* src = (const float4*)(x_select + ((size_t)b * NROW + gidx) * CDIM);
    float4*       dst = (float4*)(out + (size_t)row * CDIM);
    dst[threadIdx.x] = src[threadIdx.x]; // 64 threads x 16 B = 1 KB row
}

extern "C" void kernel_launch(void* const* d_in, const int* in_sizes, int n_in,
                              void* d_out, int out_size, void* d_ws, size_t ws_size,
                              hipStream_t stream) {
    const float* x_in     = (const float*)d_in[0];
    const float* x_select = (const float*)d_in[1];
    const float* w        = (const float*)d_in[2];
    const float* bias     = (const float*)d_in[3];
    const float* gamma    = (const float*)d_in[4];
    const float* beta     = (const float*)d_in[5];
    float* out = (float*)d_out;

    char* ws = (char*)d_ws;
    float*    scores = (float*)(ws);                      // 131072 floats (512 KB)
    float*    psum   = (float*)(ws + 524288);             // 1024 floats
    float*    psq    = (float*)(ws + 524288 + 4096);      // 1024 floats
    float*    stats  = (float*)(ws + 524288 + 8192);      // 2 floats
    unsigned* idx    = (unsigned*)(ws + 524288 + 8192 + 256); // 16384 uints

    score_wmma_kernel<<<K1_BLOCKS, 256, 0, stream>>>(x_in, w, bias, scores, psum, psq);
    reduce_kernel<<<1, 512, 0, stream>>>(psum, psq, stats);
    sort_kernel<<<BATCH, 1024, NROW * sizeof(unsigned long long), stream>>>(
        scores, stats, gamma, beta, idx);
    gather_kernel<<<BATCH * TOPN, 64, 0, stream>>>(x_select, idx, out);
}